// DPPAttention3_11304353923734
// MI455X (gfx1250) — compile-verified
//
#include <hip/hip_runtime.h>
#include <math.h>

#define BATCH 8
#define SEQ   160
#define HID   64
#define NT    10   // SEQ / 16
#define HT    4    // HID / 16

typedef __attribute__((ext_vector_type(2))) float v2f;
typedef __attribute__((ext_vector_type(8))) float v8f;

__device__ __forceinline__ v8f wmma_f32_16x16x4(v2f a, v2f b, v8f c) {
    // 8 args: (neg_a, A, neg_b, B, c_mod, C, reuse_a, reuse_b)
    return __builtin_amdgcn_wmma_f32_16x16x4_f32(false, a, false, b, (short)0, c,
                                                 false, false);
}

// ---------------------------------------------------------------------------
// Kernel 1: sampler = (X Wq^T + bq) / H^0.25 ; s2 = sampler^2 ; V = X Wv^T + bv
// One block (64 threads) per (b,l) row.
// ---------------------------------------------------------------------------
__global__ void proj_kernel(const float* __restrict__ X,
                            const float* __restrict__ Wq, const float* __restrict__ bq,
                            const float* __restrict__ Wv, const float* __restrict__ bv,
                            float* __restrict__ s2out, float* __restrict__ Vout) {
    __shared__ float xr[HID];
    int row = blockIdx.x;          // b*SEQ + l
    int h   = threadIdx.x;         // 0..63
    xr[h] = X[row * HID + h];
    __syncthreads();
    float q  = bq[h];
    float vv = bv[h];
    const float* wq = Wq + h * HID;
    const float* wv = Wv + h * HID;
#pragma unroll 8
    for (int k = 0; k < HID; ++k) {
        float xv = xr[k];
        q  = fmaf(wq[k], xv, q);
        vv = fmaf(wv[k], xv, vv);
    }
    float s = q * 0.35355339059327373f;   // 64^(-0.25)
    s2out[row * HID + h] = s * s;
    Vout[row * HID + h]  = vv;
}

// ---------------------------------------------------------------------------
// Kernel 2: Gram matrix  K[b] = s2[b] @ s2[b]^T   (160x160, contraction 64)
// One wave (32 threads) per 16x16 output tile; WMMA f32 16x16x4.
// ---------------------------------------------------------------------------
__global__ void gram_kernel(const float* __restrict__ s2, float* __restrict__ Kmat) {
    int t  = blockIdx.x;
    int b  = t / (NT * NT);
    int r  = t % (NT * NT);
    int i0 = (r / NT) * 16;
    int j0 = (r % NT) * 16;
    int lane = threadIdx.x;
    int half = lane >> 4;
    int lm   = lane & 15;
    const float* Sb = s2 + b * SEQ * HID;
    v8f acc = {};
    for (int k = 0; k < HID; k += 4) {
        int ka = k + 2 * half;
        v2f a, bb;
        a.x  = Sb[(i0 + lm) * HID + ka];       // A[m][k]   = s2[i0+m][k]
        a.y  = Sb[(i0 + lm) * HID + ka + 1];
        bb.x = Sb[(j0 + lm) * HID + ka];       // B[k][n]   = s2[j0+n][k]
        bb.y = Sb[(j0 + lm) * HID + ka + 1];
        acc = wmma_f32_16x16x4(a, bb, acc);
    }
    float* Kb = Kmat + b * SEQ * SEQ;
#pragma unroll
    for (int rr = 0; rr < 8; ++rr)
        Kb[(i0 + rr + 8 * half) * SEQ + j0 + lm] = acc[rr];
}

// ---------------------------------------------------------------------------
// Kernel 3: d[b][i] = K[b][i][i] ; S1[b] = sum_i d_i * v_i
// One block (256 threads, 160 active) per batch.
// ---------------------------------------------------------------------------
__global__ void diag_kernel(const float* __restrict__ Kmat, const float* __restrict__ mask,
                            float* __restrict__ dvec, float* __restrict__ S1) {
    __shared__ float red[256];
    int b = blockIdx.x;
    int i = threadIdx.x;
    float contrib = 0.f;
    if (i < SEQ) {
        float dv = Kmat[b * SEQ * SEQ + i * SEQ + i];
        dvec[b * SEQ + i] = dv;
        float vk = (mask[b * SEQ * SEQ + i] > -10000.0f) ? 1.f : 0.f;  // mask[b][0][i]
        contrib = dv * vk;
    }
    red[i] = contrib;
    __syncthreads();
    for (int s = 128; s > 0; s >>= 1) {
        if (i < s) red[i] += red[i + s];
        __syncthreads();
    }
    if (i == 0) S1[b] = red[0];
}

// ---------------------------------------------------------------------------
// Kernel 4: T[b] = (K[b] * v_row) @ K[b]   (160x160, contraction 160)
// One wave per 16x16 tile; WMMA f32 16x16x4.
// ---------------------------------------------------------------------------
__global__ void tmat_kernel(const float* __restrict__ Kmat, const float* __restrict__ mask,
                            float* __restrict__ T) {
    int t  = blockIdx.x;
    int b  = t / (NT * NT);
    int r  = t % (NT * NT);
    int i0 = (r / NT) * 16;
    int j0 = (r % NT) * 16;
    int lane = threadIdx.x;
    int half = lane >> 4;
    int lm   = lane & 15;
    const float* Kb = Kmat + b * SEQ * SEQ;
    const float* mb = mask + b * SEQ * SEQ;   // row 0 of attention_mask[b]
    v8f acc = {};
    for (int k = 0; k < SEQ; k += 4) {
        int ka = k + 2 * half;
        float v0 = (mb[ka]     > -10000.f) ? 1.f : 0.f;
        float v1 = (mb[ka + 1] > -10000.f) ? 1.f : 0.f;
        v2f a, bb;
        a.x  = Kb[(i0 + lm) * SEQ + ka]     * v0;  // A[m][k] = K[i0+m][k]*v[k]
        a.y  = Kb[(i0 + lm) * SEQ + ka + 1] * v1;
        bb.x = Kb[ka * SEQ + j0 + lm];             // B[k][n] = K[k][j0+n]
        bb.y = Kb[(ka + 1) * SEQ + j0 + lm];
        acc = wmma_f32_16x16x4(a, bb, acc);
    }
    float* Tb = T + b * SEQ * SEQ;
#pragma unroll
    for (int rr = 0; rr < 8; ++rr)
        Tb[(i0 + rr + 8 * half) * SEQ + j0 + lm] = acc[rr];
}

// ---------------------------------------------------------------------------
// Kernel 5: fused score + softmax.
// marginal_det[i,j] = d_i d_j S1 - d_i T_jj - K_ij K_ji S1 + K_ij T_ji
//                     + K_ji T_ij - d_j T_ii
// score = -(marg + (i==j)*d_i)*scale + mask ; softmax over j.
// One block (256 threads, 160 active) per (b,i) row.
// ---------------------------------------------------------------------------
__global__ void score_softmax_kernel(const float* __restrict__ Kmat,
                                     const float* __restrict__ T,
                                     const float* __restrict__ dvec,
                                     const float* __restrict__ S1,
                                     const float* __restrict__ mask,
                                     const float* __restrict__ scale_ptr,
                                     float* __restrict__ P) {
    __shared__ float red[256];
    int row = blockIdx.x;
    int b = row / SEQ;
    int i = row % SEQ;
    int j = threadIdx.x;
    float scale = scale_ptr[0];
    float score = -3.0e38f;
    if (j < SEQ) {
        const float* Kb = Kmat + b * SEQ * SEQ;
        const float* Tb = T    + b * SEQ * SEQ;
        float Kii = dvec[b * SEQ + i];
        float Kjj = dvec[b * SEQ + j];
        float Kij = Kb[i * SEQ + j];
        float Kji = Kb[j * SEQ + i];
        float Tii = Tb[i * SEQ + i];
        float Tjj = Tb[j * SEQ + j];
        float Tij = Tb[i * SEQ + j];
        float Tji = Tb[j * SEQ + i];
        float s1  = S1[b];
        float marg = Kii * Kjj * s1 - Kii * Tjj - Kij * Kji * s1
                   + Kij * Tji + Kji * Tij - Kjj * Tii;
        float diag = (i == j) ? Kii : 0.f;
        score = -(marg + diag) * scale + mask[b * SEQ * SEQ + i * SEQ + j];
    }
    // row max
    red[j] = score;
    __syncthreads();
    for (int s = 128; s > 0; s >>= 1) {
        if (j < s) red[j] = fmaxf(red[j], red[j + s]);
        __syncthreads();
    }
    float m = red[0];
    __syncthreads();
    float e = (j < SEQ) ? __expf(score - m) : 0.f;
    red[j] = e;
    __syncthreads();
    for (int s = 128; s > 0; s >>= 1) {
        if (j < s) red[j] += red[j + s];
        __syncthreads();
    }
    float sum = red[0];
    if (j < SEQ) P[b * SEQ * SEQ + i * SEQ + j] = e / sum;
}

// ---------------------------------------------------------------------------
// Kernel 6: context = P @ V   (160x64, contraction 160) ; WMMA tiles.
// ---------------------------------------------------------------------------
__global__ void context_kernel(const float* __restrict__ P, const float* __restrict__ V,
                               float* __restrict__ ctx) {
    int t  = blockIdx.x;
    int b  = t / (NT * HT);
    int r  = t % (NT * HT);
    int i0 = (r / HT) * 16;
    int h0 = (r % HT) * 16;
    int lane = threadIdx.x;
    int half = lane >> 4;
    int lm   = lane & 15;
    const float* Pb = P + b * SEQ * SEQ;
    const float* Vb = V + b * SEQ * HID;
    v8f acc = {};
    for (int k = 0; k < SEQ; k += 4) {
        int ka = k + 2 * half;
        v2f a, bb;
        a.x  = Pb[(i0 + lm) * SEQ + ka];       // A[m][k] = P[i0+m][k]
        a.y  = Pb[(i0 + lm) * SEQ + ka + 1];
        bb.x = Vb[ka * HID + h0 + lm];         // B[k][n] = V[k][h0+n]
        bb.y = Vb[(ka + 1) * HID + h0 + lm];
        acc = wmma_f32_16x16x4(a, bb, acc);
    }
    float* Cb = ctx + b * SEQ * HID;
#pragma unroll
    for (int rr = 0; rr < 8; ++rr)
        Cb[(i0 + rr + 8 * half) * HID + h0 + lm] = acc[rr];
}

// ---------------------------------------------------------------------------
// Kernel 7: output = ctx Wd^T + bd ; x = output + X ; LayerNorm(x)*ln_w + ln_b
// One block (64 threads) per (b,l) row.
// ---------------------------------------------------------------------------
__global__ void out_ln_kernel(const float* __restrict__ ctx,
                              const float* __restrict__ Wd, const float* __restrict__ bd,
                              const float* __restrict__ X,
                              const float* __restrict__ ln_w, const float* __restrict__ ln_b,
                              float* __restrict__ out) {
    __shared__ float cr[HID];
    __shared__ float red[HID];
    int row = blockIdx.x;
    int h   = threadIdx.x;
    cr[h] = ctx[row * HID + h];
    __syncthreads();
    float o = bd[h];
    const float* wd = Wd + h * HID;
#pragma unroll 8
    for (int k = 0; k < HID; ++k)
        o = fmaf(wd[k], cr[k], o);
    float x = o + X[row * HID + h];
    red[h] = x;
    __syncthreads();
    for (int s = 32; s > 0; s >>= 1) {
        if (h < s) red[h] += red[h + s];
        __syncthreads();
    }
    float mean = red[0] * (1.0f / HID);
    __syncthreads();
    float xc = x - mean;
    red[h] = xc * xc;
    __syncthreads();
    for (int s = 32; s > 0; s >>= 1) {
        if (h < s) red[h] += red[h + s];
        __syncthreads();
    }
    float var = red[0] * (1.0f / HID);
    float xn  = xc * rsqrtf(var + 1e-12f);
    out[row * HID + h] = ln_w[h] * xn + ln_b[h];
}

// ---------------------------------------------------------------------------
extern "C" void kernel_launch(void* const* d_in, const int* in_sizes, int n_in,
                              void* d_out, int out_size, void* d_ws, size_t ws_size,
                              hipStream_t stream) {
    const float* X     = (const float*)d_in[0];
    const float* mask  = (const float*)d_in[1];
    const float* Wq    = (const float*)d_in[2];
    const float* bq    = (const float*)d_in[3];
    const float* Wv    = (const float*)d_in[4];
    const float* bv    = (const float*)d_in[5];
    const float* Wd    = (const float*)d_in[6];
    const float* bd    = (const float*)d_in[7];
    const float* lnw   = (const float*)d_in[8];
    const float* lnb   = (const float*)d_in[9];
    const float* scale = (const float*)d_in[10];
    float* out = (float*)d_out;

    float* ws  = (float*)d_ws;
    float* s2  = ws;                         // B*L*H
    float* V   = s2 + BATCH * SEQ * HID;     // B*L*H
    float* K   = V  + BATCH * SEQ * HID;     // B*L*L
    float* T   = K  + BATCH * SEQ * SEQ;     // B*L*L
    float* P   = T  + BATCH * SEQ * SEQ;     // B*L*L
    float* C   = P  + BATCH * SEQ * SEQ;     // B*L*H
    float* dv  = C  + BATCH * SEQ * HID;     // B*L
    float* S1  = dv + BATCH * SEQ;           // B

    proj_kernel<<<BATCH * SEQ, HID, 0, stream>>>(X, Wq, bq, Wv, bv, s2, V);
    gram_kernel<<<BATCH * NT * NT, 32, 0, stream>>>(s2, K);
    diag_kernel<<<BATCH, 256, 0, stream>>>(K, mask, dv, S1);
    tmat_kernel<<<BATCH * NT * NT, 32, 0, stream>>>(K, mask, T);
    score_softmax_kernel<<<BATCH * SEQ, 256, 0, stream>>>(K, T, dv, S1, mask, scale, P);
    context_kernel<<<BATCH * NT * HT, 32, 0, stream>>>(P, V, C);
    out_ln_kernel<<<BATCH * SEQ, HID, 0, stream>>>(C, Wd, bd, X, lnw, lnb, out);
}